// GraphSAGE_29506425323820
// MI455X (gfx1250) — compile-verified
//
#include <hip/hip_runtime.h>

typedef float v2f __attribute__((ext_vector_type(2)));
typedef float v8f __attribute__((ext_vector_type(8)));

#define FDIM 64  // input & hidden feature width

// ---------------------------------------------------------------------------
// Edge scatter: one wave (32 lanes) per edge. Lane l accumulates features
// l and l+32 of feat[src[e]] into sums[dst[e]] with f32 global atomics.
// Lane 0 optionally accumulates the in-degree.
// ---------------------------------------------------------------------------
__global__ void sage_scatter(const float* __restrict__ feat,
                             const int*   __restrict__ src,
                             const int*   __restrict__ dst,
                             float*       __restrict__ sums,
                             float*       __restrict__ deg,
                             int n_edges) {
    int gid    = blockIdx.x * blockDim.x + threadIdx.x;
    int lane   = gid & 31;
    int wave   = gid >> 5;
    int nwaves = (gridDim.x * blockDim.x) >> 5;
    for (int e = wave; e < n_edges; e += nwaves) {
        int s = src[e];
        int d = dst[e];
        float v0 = feat[(size_t)s * FDIM + lane];
        float v1 = feat[(size_t)s * FDIM + lane + 32];
        atomicAdd(&sums[(size_t)d * FDIM + lane],      v0);
        atomicAdd(&sums[(size_t)d * FDIM + lane + 32], v1);
        if (deg != nullptr && lane == 0) {
            atomicAdd(&deg[d], 1.0f);
        }
    }
}

// ---------------------------------------------------------------------------
// Layer 1: h1 = relu(x @ Ws + (sum/clamp(deg,1)) @ Wn + b)   (64 -> 64)
// Block = 128 threads = 4 waves; each wave owns one 16-wide N-tile.
// Per wave: 16 x (two V_WMMA_F32_16X16X4_F32), K = 64.
//   A layout (16x4 f32): every lane row M = lane%16; lanes 16..31 hold K+2.
//   C/D layout (16x16 f32): lane col N = lane%16; VGPR r row M = r + 8*(lane>=16).
// ---------------------------------------------------------------------------
__global__ void sage_layer1(const float* __restrict__ x,
                            const float* __restrict__ sum1,
                            const float* __restrict__ deg,
                            const float* __restrict__ Ws,   // [64][64] row-major (k,n)
                            const float* __restrict__ Wn,   // [64][64]
                            const float* __restrict__ bias, // [64]
                            float*       __restrict__ h1,
                            int n_nodes) {
    int lane      = threadIdx.x & 31;
    int wv        = threadIdx.x >> 5;          // N-tile 0..3
    int node_base = blockIdx.x * 16;
    if (node_base + 16 > n_nodes) return;      // block-uniform: EXEC stays full

    int col  = lane & 15;
    int hi   = lane >> 4;                      // 0: lanes 0-15, 1: lanes 16-31
    int n0   = wv * 16;
    int arow = node_base + col;                // A-matrix row owned by this lane

    float d   = deg[arow];
    float inv = 1.0f / (d < 1.0f ? 1.0f : d);
    float b   = bias[n0 + col];
    v8f acc = {b, b, b, b, b, b, b, b};

    const float* xr = x    + (size_t)arow * FDIM;
    const float* sr = sum1 + (size_t)arow * FDIM;

#pragma unroll
    for (int i = 0; i < 16; ++i) {
        int k = 4 * i + 2 * hi;
        v2f a;  a.x  = xr[k];        a.y  = xr[k + 1];
        v2f an; an.x = sr[k] * inv;  an.y = sr[k + 1] * inv;
        v2f bs; bs.x = Ws[(size_t)k * FDIM + n0 + col];
                bs.y = Ws[(size_t)(k + 1) * FDIM + n0 + col];
        v2f bn; bn.x = Wn[(size_t)k * FDIM + n0 + col];
                bn.y = Wn[(size_t)(k + 1) * FDIM + n0 + col];
        acc = __builtin_amdgcn_wmma_f32_16x16x4_f32(false, a,  false, bs,
                                                    (short)0, acc, false, false);
        acc = __builtin_amdgcn_wmma_f32_16x16x4_f32(false, an, false, bn,
                                                    (short)0, acc, false, false);
    }

#pragma unroll
    for (int r = 0; r < 8; ++r) {
        float v = acc[r];
        v = v > 0.0f ? v : 0.0f;   // relu
        h1[(size_t)(node_base + r + 8 * hi) * FDIM + n0 + col] = v;
    }
}

// ---------------------------------------------------------------------------
// Layer 2: out = h1 @ Ws + (sum/clamp(deg,1)) @ Wn + b   (64 -> 16)
// Single 16-wide N-tile; 4 waves per block each take a different node group.
// ---------------------------------------------------------------------------
__global__ void sage_layer2(const float* __restrict__ h1,
                            const float* __restrict__ sum2,
                            const float* __restrict__ deg,
                            const float* __restrict__ Ws,   // [64][16]
                            const float* __restrict__ Wn,   // [64][16]
                            const float* __restrict__ bias, // [16]
                            float*       __restrict__ out,
                            int n_nodes) {
    int lane      = threadIdx.x & 31;
    int wv        = threadIdx.x >> 5;
    int node_base = (blockIdx.x * 4 + wv) * 16;
    if (node_base + 16 > n_nodes) return;      // wave-uniform: EXEC stays full

    int col  = lane & 15;
    int hi   = lane >> 4;
    int arow = node_base + col;

    float d   = deg[arow];
    float inv = 1.0f / (d < 1.0f ? 1.0f : d);
    float b   = bias[col];
    v8f acc = {b, b, b, b, b, b, b, b};

    const float* xr = h1   + (size_t)arow * FDIM;
    const float* sr = sum2 + (size_t)arow * FDIM;

#pragma unroll
    for (int i = 0; i < 16; ++i) {
        int k = 4 * i + 2 * hi;
        v2f a;  a.x  = xr[k];        a.y  = xr[k + 1];
        v2f an; an.x = sr[k] * inv;  an.y = sr[k + 1] * inv;
        v2f bs; bs.x = Ws[(size_t)k * 16 + col];
                bs.y = Ws[(size_t)(k + 1) * 16 + col];
        v2f bn; bn.x = Wn[(size_t)k * 16 + col];
                bn.y = Wn[(size_t)(k + 1) * 16 + col];
        acc = __builtin_amdgcn_wmma_f32_16x16x4_f32(false, a,  false, bs,
                                                    (short)0, acc, false, false);
        acc = __builtin_amdgcn_wmma_f32_16x16x4_f32(false, an, false, bn,
                                                    (short)0, acc, false, false);
    }

#pragma unroll
    for (int r = 0; r < 8; ++r) {
        out[(size_t)(node_base + r + 8 * hi) * 16 + col] = acc[r];
    }
}

// ---------------------------------------------------------------------------
// inputs: x, src, dst, W1_self, W1_neigh, b1, W2_self, W2_neigh, b2
// ---------------------------------------------------------------------------
extern "C" void kernel_launch(void* const* d_in, const int* in_sizes, int n_in,
                              void* d_out, int out_size, void* d_ws, size_t ws_size,
                              hipStream_t stream) {
    const float* x   = (const float*)d_in[0];
    const int*   src = (const int*)  d_in[1];
    const int*   dst = (const int*)  d_in[2];
    const float* W1s = (const float*)d_in[3];
    const float* W1n = (const float*)d_in[4];
    const float* b1  = (const float*)d_in[5];
    const float* W2s = (const float*)d_in[6];
    const float* W2n = (const float*)d_in[7];
    const float* b2  = (const float*)d_in[8];
    float* outp = (float*)d_out;

    int n_nodes = in_sizes[0] / FDIM;
    int n_edges = in_sizes[1];

    // Workspace layout: sum[n*64] | deg[n] | h1[n*64]   (~52 MB for n=100k)
    float* sum = (float*)d_ws;
    float* deg = sum + (size_t)n_nodes * FDIM;
    float* h1  = deg + n_nodes;

    size_t sum_bytes = (size_t)n_nodes * FDIM * sizeof(float);

    hipMemsetAsync(sum, 0, sum_bytes, stream);
    hipMemsetAsync(deg, 0, (size_t)n_nodes * sizeof(float), stream);

    // One wave per edge.
    long long threads_needed = (long long)n_edges * 32;
    int sc_blocks = (int)((threads_needed + 255) / 256);

    // Layer 1: aggregate x, then fused GEMM+bias+relu.
    sage_scatter<<<sc_blocks, 256, 0, stream>>>(x, src, dst, sum, deg, n_edges);
    int l1_blocks = (n_nodes + 15) / 16;
    sage_layer1<<<l1_blocks, 128, 0, stream>>>(x, sum, deg, W1s, W1n, b1, h1, n_nodes);

    // Layer 2: re-zero sums, aggregate h1 (degree unchanged), fused GEMM+bias.
    hipMemsetAsync(sum, 0, sum_bytes, stream);
    sage_scatter<<<sc_blocks, 256, 0, stream>>>(h1, src, dst, sum, nullptr, n_edges);
    int l2_blocks = (n_nodes + 63) / 64;
    sage_layer2<<<l2_blocks, 128, 0, stream>>>(h1, sum, deg, W2s, W2n, b2, outp, n_nodes);
}